// DeformConv2d_28870770163856
// MI455X (gfx1250) — compile-verified
//
#include <hip/hip_runtime.h>

typedef __attribute__((ext_vector_type(2))) float v2f;
typedef __attribute__((ext_vector_type(8))) float v8f;

#define KH 3
#define KW 3
#define NPTS 9          // KH*KW
#define CIN 64
#define COUT 18         // 2*NPTS
#define Bsz 4
#define H 128
#define W 128
#define KTOT (KH*KW*CIN)          // 576
#define NPIX (Bsz*H*W)            // 65536
#define LDSK_ELEMS (KTOT*COUT + 32)  // 10400 floats (pad lets junk columns read in-range)

// ---------------------------------------------------------------------------
// Kernel 1: offset-generating conv as im2col GEMM with V_WMMA_F32_16X16X4_F32.
// M = NPIX pixels, N = 18 (padded to 32, junk cols never stored), K = 576.
// 256 threads = 8 waves; wave (w) -> M-tile = blockIdx*4 + w/2, N-tile = w%2.
// ---------------------------------------------------------------------------
__global__ __launch_bounds__(256)
void deform_offset_conv_wmma(const float* __restrict__ in,
                             const float* __restrict__ kern,
                             float* __restrict__ offs) {
    __shared__ float ldsK[LDSK_ELEMS];

    // Stage the kernel weights: flat (kh,kw,c,n) layout == k*18 + n already.
    for (int idx = threadIdx.x; idx < LDSK_ELEMS; idx += 256)
        ldsK[idx] = (idx < KTOT * COUT) ? kern[idx] : 0.0f;
    __syncthreads();

    const int lane  = threadIdx.x & 31;
    const int wave  = threadIdx.x >> 5;
    const int mtile = blockIdx.x * 4 + (wave >> 1);
    const int ntile = wave & 1;

    const int mrow  = lane & 15;     // A-matrix row / D column-lane
    const int khalf = lane >> 4;     // 0: K=0,1  1: K=2,3 within each 4-step

    const int m  = mtile * 16 + mrow;    // pixel row of A this lane feeds
    const int b_ = m >> 14;              // m / (128*128)
    const int i_ = (m >> 7) & 127;
    const int j_ = m & 127;
    const int n_ = ntile * 16 + mrow;    // B column this lane reads (may be junk >=18)

    v8f acc = {0.f, 0.f, 0.f, 0.f, 0.f, 0.f, 0.f, 0.f};

    for (int kh = 0; kh < KH; ++kh) {
        const int pi = i_ + kh - 1;
        for (int kw = 0; kw < KW; ++kw) {
            const int pj  = j_ + kw - 1;
            const bool inb = ((unsigned)pi < (unsigned)H) && ((unsigned)pj < (unsigned)W);
            const int spi = inb ? pi : 0;
            const int spj = inb ? pj : 0;
            const float* arow = in + ((((b_ * H + spi) * W + spj)) << 6);
            const int kb = (kh * KW + kw) << 6;   // base K of this tap
            for (int c0 = 0; c0 < CIN; c0 += 4) {
                const int cc = c0 + 2 * khalf;    // even -> 8B aligned
                const float2 av = inb ? *(const float2*)(arow + cc)
                                      : make_float2(0.0f, 0.0f);
                v2f a; a.x = av.x; a.y = av.y;
                const int kk = kb + cc;
                v2f bb;
                bb.x = ldsK[kk * COUT + n_];
                bb.y = ldsK[kk * COUT + COUT + n_];
                acc = __builtin_amdgcn_wmma_f32_16x16x4_f32(
                    false, a, false, bb, (short)0, acc, false, false);
            }
        }
    }

    // D layout: VGPR r, lanes 0-15 -> M=r, lanes 16-31 -> M=8+r; N = lane%16.
    const int nb = ntile * 16 + (lane & 15);
    if (nb < COUT) {
        const int mBase = mtile * 16 + 8 * khalf;
        #pragma unroll
        for (int r = 0; r < 8; ++r)
            offs[(mBase + r) * COUT + nb] = acc[r];
    }
}

// ---------------------------------------------------------------------------
// Kernel 2: bilinear gather + blend. 16 threads per (pixel, sample point),
// one float4 (4 channels) per thread; contiguous 256B corner reads, coalesced
// 16B stores. Replicates the keras quirks exactly (p0x = j+1, p0y = i+1,
// x-channels index spatial axis 0 of the padded input).
// ---------------------------------------------------------------------------
__global__ __launch_bounds__(256)
void deform_sample(const float* __restrict__ in,
                   const float* __restrict__ offs,
                   float* __restrict__ out) {
    const int tid = blockIdx.x * 256 + threadIdx.x;
    const int ps  = tid >> 4;          // pixel-sample id
    const int cq  = tid & 15;          // channel quad (4 floats)

    const int n   = ps % NPTS;
    const int pix = ps / NPTS;
    const int j   = pix & 127;
    const int i   = (pix >> 7) & 127;
    const int b   = pix >> 14;

    const float ox = offs[pix * COUT + n];
    const float oy = offs[pix * COUT + NPTS + n];

    // p = p0 + pn + offset, in padded (130x130) coordinates
    float px = (float)(j + 1) + (float)(n % 3 - 1) + ox;
    float py = (float)(i + 1) + (float)(n / 3 - 1) + oy;
    px = fminf(fmaxf(px, 0.0f), 129.0f);
    py = fminf(fmaxf(py, 0.0f), 129.0f);

    const float lbx = floorf(px), rtx = ceilf(px);
    const float lby = floorf(py), rty = ceilf(py);

    const float wx0 = rtx - px, wx1 = px - lbx;
    const float wy0 = rty - py, wy1 = py - lby;
    const float g_rt = wx0 * wy0;   // pairs with X(lbx,lby)
    const float g_lt = wx1 * wy0;   // pairs with X(rtx,lby)
    const float g_rb = wx0 * wy1;   // pairs with X(lbx,rty)
    const float g_lb = wx1 * wy1;   // pairs with X(rtx,rty)

    const int ulb = (int)lbx, urt = (int)rtx;
    const int vlb = (int)lby, vrt = (int)rty;

    const int cbase = cq << 2;
    auto corner = [&](int u, int v) -> float4 {
        const bool inb = (u >= 1) && (u <= H) && (v >= 1) && (v <= W);
        if (!inb) return make_float4(0.f, 0.f, 0.f, 0.f);
        return *(const float4*)(in + (((b * H + (u - 1)) * W + (v - 1)) << 6) + cbase);
    };

    const float4 x00 = corner(ulb, vlb);
    const float4 x10 = corner(urt, vlb);
    const float4 x01 = corner(ulb, vrt);
    const float4 x11 = corner(urt, vrt);

    float4 val;
    val.x = g_rt * x00.x + g_lt * x10.x + g_rb * x01.x + g_lb * x11.x;
    val.y = g_rt * x00.y + g_lt * x10.y + g_rb * x01.y + g_lb * x11.y;
    val.z = g_rt * x00.z + g_lt * x10.z + g_rb * x01.z + g_lb * x11.z;
    val.w = g_rt * x00.w + g_lt * x10.w + g_rb * x01.w + g_lb * x11.w;

    // (B, Ho, Wo, N, C) -> (B, Ho*3, Wo*3, C)
    const int orow = i * 3 + n / 3;
    const int ocol = j * 3 + n % 3;
    float* dst = out + (((b * (H * 3) + orow) * (W * 3) + ocol) << 6) + cbase;
    *(float4*)dst = val;
}

extern "C" void kernel_launch(void* const* d_in, const int* in_sizes, int n_in,
                              void* d_out, int out_size, void* d_ws, size_t ws_size,
                              hipStream_t stream) {
    const float* in   = (const float*)d_in[0];   // (4,128,128,64) f32
    const float* kern = (const float*)d_in[1];   // (3,3,64,18)   f32
    float* out  = (float*)d_out;                 // (4,384,384,64) f32
    float* offs = (float*)d_ws;                  // (4,128,128,18) f32 scratch

    // Kernel 1: 4096 M-tiles / 4 per block = 1024 blocks of 256 threads.
    deform_offset_conv_wmma<<<NPIX / 64, 256, 0, stream>>>(in, kern, offs);

    // Kernel 2: 65536*9 pixel-samples * 16 threads = 9,437,184 threads.
    const int total = NPIX * NPTS * 16;
    deform_sample<<<total / 256, 256, 0, stream>>>(in, offs, out);

    (void)in_sizes; (void)n_in; (void)out_size; (void)ws_size;
}